// CTAN_24077586661674
// MI455X (gfx1250) — compile-verified
//
#include <hip/hip_runtime.h>
#include <hip/hip_bf16.h>

// Problem constants (fixed by the reference)
#define NB        50000     // nodes in batch (divisible by 16: 3125 tiles)
#define EB        600000    // edges (divisible by 16: 37500 tiles; by 8: 75000)
#define MEMD      128
#define EDGED     72
#define TIMED     56
#define N_ITERS   3
#define EPSV      0.1f
#define GAMMAV    0.1f
#define INV_SQRT_D 0.08838834764831845f  // 1/sqrt(128)

typedef __attribute__((ext_vector_type(2))) float v2f;
typedef __attribute__((ext_vector_type(8))) float v8f;

// ---- ordered-uint encoding of float for atomic segment-max ----
__device__ __forceinline__ unsigned f2ord(float f) {
  unsigned u = __float_as_uint(f);
  return (u & 0x80000000u) ? ~u : (u | 0x80000000u);
}
__device__ __forceinline__ float ord2f(unsigned u) {
  unsigned b = (u & 0x80000000u) ? (u & 0x7fffffffu) : ~u;
  return __uint_as_float(b);
}

// ---- A_eff = aW - aW^T - gamma*I --------------------------------------------
__global__ __launch_bounds__(256) void build_A_kernel(const float* __restrict__ aW,
                                                      float* __restrict__ A) {
  int i = blockIdx.x * 256 + threadIdx.x;   // 16384 total
  int r = i >> 7, c = i & 127;
  A[i] = aW[i] - aW[c * 128 + r] - ((r == c) ? GAMMAV : 0.0f);
}

// ---- encoder: x = concat(memory[n_id], static[n_id]) @ enc_w^T + enc_b ------
// block = 256 (8 waves); each block: 16 nodes; each wave: one 16x16 out tile
__global__ __launch_bounds__(256) void enc_kernel(const float* __restrict__ memf,
                                                  const float* __restrict__ statf,
                                                  const int*   __restrict__ n_id,
                                                  const float* __restrict__ enc_w,
                                                  const float* __restrict__ enc_b,
                                                  float* __restrict__ x) {
  __shared__ float zt[16 * 256];            // 16 KB: z tile [16 rows x 256 K]
  const int row0 = blockIdx.x * 16;
  const int tid  = threadIdx.x;
  // stage z tile: thread -> (row = tid/16, 16-float segment = tid%16)
  {
    int r = tid >> 4, seg = tid & 15;
    int nid = n_id[row0 + r];
    const float* src = (seg < 8) ? (memf + (size_t)nid * 128 + seg * 16)
                                 : (statf + (size_t)nid * 128 + (seg - 8) * 16);
    const float4* sp = (const float4*)src;
    float4* dp = (float4*)(zt + r * 256 + seg * 16);
    dp[0] = sp[0]; dp[1] = sp[1]; dp[2] = sp[2]; dp[3] = sp[3];
  }
  __syncthreads();
  const int wave = tid >> 5, lane = tid & 31;
  const int ml = lane & 15, hl = lane >> 4;
  const int n0 = wave * 16;
  const float* wrow = enc_w + (size_t)(n0 + ml) * 256;  // B[k][n] = enc_w[n][k]
  v8f c = {};
  #pragma unroll 8
  for (int k0 = 0; k0 < 256; k0 += 4) {
    int kA = k0 + hl * 2;
    v2f a = *(const v2f*)(zt + ml * 256 + kA);
    v2f b = *(const v2f*)(wrow + kA);
    c = __builtin_amdgcn_wmma_f32_16x16x4_f32(false, a, false, b, (short)0, c,
                                              false, false);
  }
  float bias = enc_b[n0 + ml];
  #pragma unroll
  for (int rI = 0; rI < 8; ++rI) {
    int M = rI + hl * 8;
    x[(size_t)(row0 + M) * 128 + n0 + ml] = c[rI] + bias;
  }
}

// ---- edge features: e = [msg || cos(dt*time_w+time_b)] @ we^T  (bf16 out) ---
__global__ __launch_bounds__(256) void edge_feat_kernel(const int*   __restrict__ srcI,
                                                        const int*   __restrict__ tt,
                                                        const float* __restrict__ msg,
                                                        const int*   __restrict__ n_id,
                                                        const int*   __restrict__ lastup,
                                                        const float* __restrict__ time_w,
                                                        const float* __restrict__ time_b,
                                                        const float* __restrict__ we,
                                                        __hip_bfloat16* __restrict__ ebuf) {
  __shared__ float at[16 * 128];            // edge_attr tile, 8 KB
  __shared__ float relt[16];
  const int e0  = blockIdx.x * 16;
  const int tid = threadIdx.x;
  if (tid < 16) {
    int ee = e0 + tid;
    int dt = lastup[n_id[srcI[ee]]] - tt[ee];
    relt[tid] = fabsf((float)dt);
  }
  for (int i = tid; i < 16 * EDGED; i += 256) {
    int rr = i / EDGED, cc = i % EDGED;
    at[rr * 128 + cc] = msg[(size_t)(e0 + rr) * EDGED + cc];
  }
  __syncthreads();
  for (int i = tid; i < 16 * TIMED; i += 256) {
    int rr = i / TIMED, cc = i % TIMED;
    at[rr * 128 + EDGED + cc] = cosf(relt[rr] * time_w[cc] + time_b[cc]);
  }
  __syncthreads();
  const int wave = tid >> 5, lane = tid & 31;
  const int ml = lane & 15, hl = lane >> 4;
  const int n0 = wave * 16;
  const float* wrow = we + (size_t)(n0 + ml) * 128;
  v8f c = {};
  #pragma unroll 8
  for (int k0 = 0; k0 < 128; k0 += 4) {
    int kA = k0 + hl * 2;
    v2f a = *(const v2f*)(at + ml * 128 + kA);
    v2f b = *(const v2f*)(wrow + kA);
    c = __builtin_amdgcn_wmma_f32_16x16x4_f32(false, a, false, b, (short)0, c,
                                              false, false);
  }
  #pragma unroll
  for (int rI = 0; rI < 8; ++rI) {
    int M = rI + hl * 8;
    ebuf[(size_t)(e0 + M) * 128 + n0 + ml] = __float2bfloat16(c[rI]);
  }
}

// ---- generic node GEMM: out = x @ W^T + b  (used for q,k,v) -----------------
__global__ __launch_bounds__(256) void qkv_kernel(const float* __restrict__ xg,
                                                  const float* __restrict__ W,
                                                  const float* __restrict__ bvec,
                                                  float* __restrict__ out) {
  __shared__ float zt[16 * 128];            // 8 KB
  const int row0 = blockIdx.x * 16;
  const int tid  = threadIdx.x;
  {
    int r = tid >> 4, seg = tid & 15;       // 8 floats per thread
    const float4* sp = (const float4*)(xg + (size_t)(row0 + r) * 128 + seg * 8);
    float4* dp = (float4*)(zt + r * 128 + seg * 8);
    dp[0] = sp[0]; dp[1] = sp[1];
  }
  __syncthreads();
  const int wave = tid >> 5, lane = tid & 31;
  const int ml = lane & 15, hl = lane >> 4;
  const int n0 = wave * 16;
  const float* wrow = W + (size_t)(n0 + ml) * 128;
  v8f c = {};
  #pragma unroll 8
  for (int k0 = 0; k0 < 128; k0 += 4) {
    int kA = k0 + hl * 2;
    v2f a = *(const v2f*)(zt + ml * 128 + kA);
    v2f b = *(const v2f*)(wrow + kA);
    c = __builtin_amdgcn_wmma_f32_16x16x4_f32(false, a, false, b, (short)0, c,
                                              false, false);
  }
  float bias = bvec[n0 + ml];
  #pragma unroll
  for (int rI = 0; rI < 8; ++rI) {
    int M = rI + hl * 8;
    out[(size_t)(row0 + M) * 128 + n0 + ml] = c[rI] + bias;
  }
}

// ---- zero h[N*128], s[N], m_u[N] -------------------------------------------
__global__ __launch_bounds__(256) void init_kernel(float* __restrict__ h,
                                                   float* __restrict__ s,
                                                   unsigned* __restrict__ m_u) {
  long i = (long)blockIdx.x * 256 + threadIdx.x;
  long hn = (long)NB * 128;
  if (i < hn)                { h[i] = 0.0f; }
  else if (i < hn + NB)      { s[i - hn] = 0.0f; }
  else if (i < hn + 2 * NB)  { m_u[i - hn - NB] = 0u; }  // below ord of any finite float
}

// ---- alpha[e] = inv_sqrt_d * dot(q[dst], k[src]+e); segment-max via atomics -
__global__ __launch_bounds__(256) void alpha_kernel(const float* __restrict__ q,
                                                    const float* __restrict__ k,
                                                    const __hip_bfloat16* __restrict__ ebuf,
                                                    const int* __restrict__ srcI,
                                                    const int* __restrict__ dstI,
                                                    float* __restrict__ alpha,
                                                    unsigned* __restrict__ m_u) {
  const int wave = threadIdx.x >> 5, lane = threadIdx.x & 31;
  const int e = blockIdx.x * 8 + wave;      // exact: EB/8 blocks
  const int s = srcI[e], d = dstI[e];
  const float4* qp = (const float4*)(q + (size_t)d * 128);
  const float4* kp = (const float4*)(k + (size_t)s * 128);
  const __hip_bfloat16* ep = ebuf + (size_t)e * 128 + lane * 4;
  float4 qv = qp[lane];
  float4 kv = kp[lane];
  float e0 = __bfloat162float(ep[0]), e1 = __bfloat162float(ep[1]);
  float e2 = __bfloat162float(ep[2]), e3 = __bfloat162float(ep[3]);
  float acc = qv.x * (kv.x + e0) + qv.y * (kv.y + e1) +
              qv.z * (kv.z + e2) + qv.w * (kv.w + e3);
  #pragma unroll
  for (int off = 16; off > 0; off >>= 1) acc += __shfl_down(acc, off, 32);
  if (lane == 0) {
    float al = acc * INV_SQRT_D;
    alpha[e] = al;
    atomicMax(&m_u[d], f2ord(al));
  }
}

// ---- ex = exp(alpha - m[dst]); s[dst] += ex (ex overwrites alpha) -----------
__global__ __launch_bounds__(256) void exp_kernel(float* __restrict__ alpha,
                                                  const int* __restrict__ dstI,
                                                  const unsigned* __restrict__ m_u,
                                                  float* __restrict__ s) {
  int e = blockIdx.x * 256 + threadIdx.x;
  if (e >= EB) return;
  int d = dstI[e];
  float ex = __expf(alpha[e] - ord2f(m_u[d]));
  alpha[e] = ex;
  atomicAdd(&s[d], ex);
}

// ---- h[dst] += (v[src] + e) * (ex / s[dst]) --------------------------------
__global__ __launch_bounds__(256) void accum_kernel(const float* __restrict__ v,
                                                    const __hip_bfloat16* __restrict__ ebuf,
                                                    const float* __restrict__ alpha,
                                                    const float* __restrict__ s,
                                                    const int* __restrict__ srcI,
                                                    const int* __restrict__ dstI,
                                                    float* __restrict__ h) {
  const int wave = threadIdx.x >> 5, lane = threadIdx.x & 31;
  const int e = blockIdx.x * 8 + wave;
  const int sn = srcI[e], d = dstI[e];
  const float a = alpha[e] / s[d];
  const float4* vp = (const float4*)(v + (size_t)sn * 128);
  const __hip_bfloat16* ep = ebuf + (size_t)e * 128 + lane * 4;
  float4 vv = vp[lane];
  float* hp = h + (size_t)d * 128 + lane * 4;
  atomicAdd(hp + 0, (vv.x + __bfloat162float(ep[0])) * a);
  atomicAdd(hp + 1, (vv.y + __bfloat162float(ep[1])) * a);
  atomicAdd(hp + 2, (vv.z + __bfloat162float(ep[2])) * a);
  atomicAdd(hp + 3, (vv.w + __bfloat162float(ep[3])) * a);
}

// ---- x = x + eps * tanh(x @ A_eff^T + h + abias)  (in-place, per-row tiles) -
__global__ __launch_bounds__(256) void update_kernel(float* __restrict__ x,
                                                     const float* __restrict__ A,
                                                     const float* __restrict__ h,
                                                     const float* __restrict__ abias) {
  __shared__ float zt[16 * 128];
  const int row0 = blockIdx.x * 16;
  const int tid  = threadIdx.x;
  {
    int r = tid >> 4, seg = tid & 15;
    const float4* sp = (const float4*)(x + (size_t)(row0 + r) * 128 + seg * 8);
    float4* dp = (float4*)(zt + r * 128 + seg * 8);
    dp[0] = sp[0]; dp[1] = sp[1];
  }
  __syncthreads();
  const int wave = tid >> 5, lane = tid & 31;
  const int ml = lane & 15, hl = lane >> 4;
  const int n0 = wave * 16;
  const float* arow = A + (size_t)(n0 + ml) * 128;
  v8f c = {};
  #pragma unroll 8
  for (int k0 = 0; k0 < 128; k0 += 4) {
    int kA = k0 + hl * 2;
    v2f a = *(const v2f*)(zt + ml * 128 + kA);
    v2f b = *(const v2f*)(arow + kA);
    c = __builtin_amdgcn_wmma_f32_16x16x4_f32(false, a, false, b, (short)0, c,
                                              false, false);
  }
  float bias = abias[n0 + ml];
  #pragma unroll
  for (int rI = 0; rI < 8; ++rI) {
    int M = rI + hl * 8;
    size_t idx = (size_t)(row0 + M) * 128 + n0 + ml;
    float hv = tanhf(c[rI] + h[idx] + bias);
    x[idx] = zt[M * 128 + n0 + ml] + EPSV * hv;
  }
}

extern "C" void kernel_launch(void* const* d_in, const int* in_sizes, int n_in,
                              void* d_out, int out_size, void* d_ws, size_t ws_size,
                              hipStream_t stream) {
  const int*   n_id   = (const int*)d_in[0];
  const int*   eidx   = (const int*)d_in[1];
  const int*   tt     = (const int*)d_in[2];
  const float* msg    = (const float*)d_in[3];
  const float* statf  = (const float*)d_in[4];
  const float* memf   = (const float*)d_in[5];
  const int*   lastup = (const int*)d_in[6];
  const float* enc_w  = (const float*)d_in[7];
  const float* enc_b  = (const float*)d_in[8];
  const float* time_w = (const float*)d_in[9];
  const float* time_b = (const float*)d_in[10];
  const float* wq = (const float*)d_in[11]; const float* bq = (const float*)d_in[12];
  const float* wk = (const float*)d_in[13]; const float* bk = (const float*)d_in[14];
  const float* wv = (const float*)d_in[15]; const float* bv = (const float*)d_in[16];
  const float* we = (const float*)d_in[17];
  const float* aW = (const float*)d_in[18]; const float* abias = (const float*)d_in[19];

  float* x = (float*)d_out;                 // x lives in d_out throughout
  const int* srcI = eidx;
  const int* dstI = eidx + EB;

  // workspace carve-up (~259 MB total)
  char* p = (char*)d_ws;
  float*    A_eff = (float*)p;          p += (size_t)128 * 128 * 4;
  float*    qb    = (float*)p;          p += (size_t)NB * 128 * 4;
  float*    kb    = (float*)p;          p += (size_t)NB * 128 * 4;
  float*    vb    = (float*)p;          p += (size_t)NB * 128 * 4;
  float*    hb    = (float*)p;          p += (size_t)NB * 128 * 4;
  float*    al    = (float*)p;          p += (size_t)EB * 4;
  float*    sb    = (float*)p;          p += (size_t)NB * 4;
  unsigned* mb    = (unsigned*)p;       p += (size_t)NB * 4;
  __hip_bfloat16* ebuf = (__hip_bfloat16*)p;  // EB*128*2 bytes

  build_A_kernel<<<64, 256, 0, stream>>>(aW, A_eff);
  enc_kernel<<<NB / 16, 256, 0, stream>>>(memf, statf, n_id, enc_w, enc_b, x);
  edge_feat_kernel<<<EB / 16, 256, 0, stream>>>(srcI, tt, msg, n_id, lastup,
                                                time_w, time_b, we, ebuf);
  for (int it = 0; it < N_ITERS; ++it) {
    qkv_kernel<<<NB / 16, 256, 0, stream>>>(x, wq, bq, qb);
    qkv_kernel<<<NB / 16, 256, 0, stream>>>(x, wk, bk, kb);
    qkv_kernel<<<NB / 16, 256, 0, stream>>>(x, wv, bv, vb);
    init_kernel<<<((long)NB * 130 + 255) / 256, 256, 0, stream>>>(hb, sb, mb);
    alpha_kernel<<<EB / 8, 256, 0, stream>>>(qb, kb, ebuf, srcI, dstI, al, mb);
    exp_kernel<<<(EB + 255) / 256, 256, 0, stream>>>(al, dstI, mb, sb);
    accum_kernel<<<EB / 8, 256, 0, stream>>>(vb, ebuf, al, sb, srcI, dstI, hb);
    update_kernel<<<NB / 16, 256, 0, stream>>>(x, A_eff, hb, abias);
  }
}